// HadamardTransform_72713796322122
// MI455X (gfx1250) — compile-verified
//
#include <hip/hip_runtime.h>
#include <hip/hip_bf16.h>

// FWHT-4096 on 8192 rows via Y = H64 * M * H64  (M = row viewed as 64x64),
// H64[r][c] = (-1)^popc(r&c), using V_WMMA_F32_16X16X4_F32 chains (K=4 x16).
//
// Both matmul passes are computed as "H * (something)" so the procedurally
// generated H fragments (A operand) are built ONCE per wave and reused:
//   pass 1:  T  = H * M        (B = M from LDS,  store T^T to LDS)
//   pass 2:  Y^T = H * T^T     (B = T^T from LDS, store (Y^T)^T = Y to LDS)
// Epilogues write transposed tiles -> each lane writes 8 contiguous floats
// (two aligned float4 LDS stores). Stride-68 padding keeps column-pair reads
// bank-conflict-free (bank = (4*row + col) mod 64).

typedef float v2f __attribute__((ext_vector_type(2)));
typedef float v8f __attribute__((ext_vector_type(8)));

#define FWHT_N     4096
#define FWHT_DIM   64
#define LDS_STRIDE 68   // floats; %4==0 keeps float4 alignment in padded rows

static __device__ __forceinline__ float hsign(int r, int c) {
  return (__popc(r & c) & 1) ? -1.0f : 1.0f;
}

__global__ __launch_bounds__(128, 1)
void fwht4096_wmma(const float* __restrict__ x, float* __restrict__ y, int nrows) {
  __shared__ __align__(16) float bufM[FWHT_DIM * LDS_STRIDE];
  __shared__ __align__(16) float bufT[FWHT_DIM * LDS_STRIDE];

  const int row = blockIdx.x;
  if (row >= nrows) return;

  const int tid  = threadIdx.x;   // 0..127
  const int wave = tid >> 5;      // 0..3 : tile-row of each 64x64 product
  const int lane = tid & 31;
  const int half = lane >> 4;     // 0/1 half-wave
  const int l16  = lane & 15;

  // ---- coalesced load: global float4 -> padded LDS (row-major M) ----
  {
    const float4* src = (const float4*)(x + (size_t)row * FWHT_N);
#pragma unroll
    for (int it = 0; it < 8; ++it) {
      const int idx4 = tid + it * 128;       // float4 index 0..1023
      const float4 v = src[idx4];
      const int e = idx4 << 2;
      const int r = e >> 6, c = e & 63;      // float4 never crosses a row
      *(float4*)&bufM[r * LDS_STRIDE + c] = v;
    }
  }
  __syncthreads();

  // ---- H A-fragments for this wave's tile-row, built once, used twice ----
  // A chunk (k,q): lanes 0-15 hold K = 16k+4q+{0,1}, lanes 16-31 K = +{2,3}
  v2f Ha[4][4];
#pragma unroll
  for (int k = 0; k < 4; ++k) {
#pragma unroll
    for (int q = 0; q < 4; ++q) {
      const int rr = 16 * wave + l16;
      const int cc = 16 * k + 4 * q + 2 * half;
      Ha[k][q][0] = hsign(rr, cc);
      Ha[k][q][1] = hsign(rr, cc + 1);
    }
  }

  const int bc = l16;  // per-lane N column within a tile

  // ---- pass 1: T = H * M ; store T transposed into bufT ----
#pragma unroll
  for (int j = 0; j < 4; ++j) {
    v8f acc = {};
#pragma unroll
    for (int k = 0; k < 4; ++k) {
#pragma unroll
      for (int q = 0; q < 4; ++q) {
        const int br = 16 * k + 4 * q + 2 * half;      // B K-row
        v2f b;
        b[0] = bufM[br * LDS_STRIDE + 16 * j + bc];
        b[1] = bufM[(br + 1) * LDS_STRIDE + 16 * j + bc];
        acc = __builtin_amdgcn_wmma_f32_16x16x4_f32(
            false, Ha[k][q], false, b, (short)0, acc, false, false);
      }
    }
    // D element (m=i+8*half, n=l16) = T[16w+m][16j+n]; store T^T:
    // bufT[(16j+l16)*S + 16w+8half + i] -> 8 contiguous floats per lane.
    float* p = &bufT[(16 * j + l16) * LDS_STRIDE + 16 * wave + 8 * half];
    float4 lo, hi;
    lo.x = acc[0]; lo.y = acc[1]; lo.z = acc[2]; lo.w = acc[3];
    hi.x = acc[4]; hi.y = acc[5]; hi.z = acc[6]; hi.w = acc[7];
    *(float4*)p = lo;
    *(float4*)(p + 4) = hi;
  }
  __syncthreads();

  // ---- pass 2: Z = Y^T = H * T^T ; store Z^T = Y (row-major) into bufM ----
#pragma unroll
  for (int j = 0; j < 4; ++j) {
    v8f acc = {};
#pragma unroll
    for (int k = 0; k < 4; ++k) {
#pragma unroll
      for (int q = 0; q < 4; ++q) {
        const int br = 16 * k + 4 * q + 2 * half;      // B K-row of T^T
        v2f b;
        b[0] = bufT[br * LDS_STRIDE + 16 * j + bc];
        b[1] = bufT[(br + 1) * LDS_STRIDE + 16 * j + bc];
        acc = __builtin_amdgcn_wmma_f32_16x16x4_f32(
            false, Ha[k][q], false, b, (short)0, acc, false, false);
      }
    }
    // D element (m,n) = Z[16w+m][16j+n] = Y[16j+n][16w+m]; store Y row-major.
    float* p = &bufM[(16 * j + l16) * LDS_STRIDE + 16 * wave + 8 * half];
    float4 lo, hi;
    lo.x = acc[0] * 0.015625f; lo.y = acc[1] * 0.015625f;  // 1/64 exact
    lo.z = acc[2] * 0.015625f; lo.w = acc[3] * 0.015625f;
    hi.x = acc[4] * 0.015625f; hi.y = acc[5] * 0.015625f;
    hi.z = acc[6] * 0.015625f; hi.w = acc[7] * 0.015625f;
    *(float4*)p = lo;
    *(float4*)(p + 4) = hi;
  }
  __syncthreads();

  // ---- coalesced store: padded LDS -> global float4 ----
  {
    float4* dst = (float4*)(y + (size_t)row * FWHT_N);
#pragma unroll
    for (int it = 0; it < 8; ++it) {
      const int idx4 = tid + it * 128;
      const int e = idx4 << 2;
      const int r = e >> 6, c = e & 63;
      dst[idx4] = *(const float4*)&bufM[r * LDS_STRIDE + c];
    }
  }
}

extern "C" void kernel_launch(void* const* d_in, const int* in_sizes, int n_in,
                              void* d_out, int out_size, void* d_ws, size_t ws_size,
                              hipStream_t stream) {
  (void)n_in; (void)d_ws; (void)ws_size; (void)out_size;
  const float* x = (const float*)d_in[0];
  float* y = (float*)d_out;
  const int nrows = in_sizes[0] / FWHT_N;   // 8192
  fwht4096_wmma<<<dim3(nrows), dim3(128), 0, stream>>>(x, y, nrows);
}